// BatchAllTripletLoss_87222195847820
// MI455X (gfx1250) — compile-verified
//
#include <hip/hip_runtime.h>
#include <hip/hip_bf16.h>

#define MARGIN 0.1f
#define EPS_F 1e-16f

typedef __attribute__((ext_vector_type(2))) float v2f;
typedef __attribute__((ext_vector_type(8))) float v8f;

// ---------------------------------------------------------------------------
// Kernel 0: zero the scalar accumulators (must run every launch: harness does
// not re-poison between replays and kernel_launch must be deterministic).
// ---------------------------------------------------------------------------
__global__ void init_acc_kernel(float* sum_acc, unsigned int* cnt_acc) {
    if (threadIdx.x == 0) {
        *sum_acc = 0.0f;
        *cnt_acc = 0u;
    }
}

// ---------------------------------------------------------------------------
// Kernel 1: squared row norms  sq[i] = sum_k E[i,k]^2  (== diag of Gram)
// ---------------------------------------------------------------------------
__global__ void rownorm_kernel(const float* __restrict__ E,
                               float* __restrict__ sq, int N, int D) {
    int i = blockIdx.x * blockDim.x + threadIdx.x;
    if (i >= N) return;
    const float4* row = (const float4*)(E + (size_t)i * D);
    float s = 0.0f;
    for (int k = 0; k < D / 4; ++k) {
        float4 v = row[k];
        s += v.x * v.x + v.y * v.y + v.z * v.z + v.w * v.w;
    }
    sq[i] = s;
}

// ---------------------------------------------------------------------------
// Kernel 2: one wave (32 lanes) computes a 16x16 tile of G = E * E^T via
// V_WMMA_F32_16X16X4_F32, then applies the distance epilogue
//   d = sqrt(max(sq_i - 2g + sq_j, 0) + eps*iszero) * (1 - iszero)
//
// A-matrix 16x4 f32 layout (ISA 7.12.2): lane L holds row M=L&15;
//   lanes 0-15:  VGPR0=K0, VGPR1=K1 ; lanes 16-31: VGPR0=K2, VGPR1=K3.
// B-matrix 4x16 is the mirrored pattern, so for B = E^T both operands are
// loaded with the identical {row = base + (L&15), col = k0 + 2*(L>>4)}
// 8-byte global_load_b64 pattern.
// ---------------------------------------------------------------------------
__global__ __launch_bounds__(32) void gram_pd_kernel(
        const float* __restrict__ E, const float* __restrict__ sq,
        float* __restrict__ pd, int N, int D) {
    const int lane  = threadIdx.x;            // 0..31, wave32
    const int ib    = blockIdx.y * 16;        // tile row base (anchor i)
    const int jb    = blockIdx.x * 16;        // tile col base (j)
    const int rsub  = lane & 15;
    const int khalf = (lane >> 4) << 1;       // 0 or 2

    const float* arow = E + (size_t)(ib + rsub) * D + khalf;
    const float* brow = E + (size_t)(jb + rsub) * D + khalf;

    v8f c = {};
    for (int k0 = 0; k0 < D; k0 += 4) {
        v2f a = *(const v2f*)(arow + k0);
        v2f b = *(const v2f*)(brow + k0);
        // (neg_a, A, neg_b, B, c_mod, C, reuse_a, reuse_b)
        c = __builtin_amdgcn_wmma_f32_16x16x4_f32(
                false, a, false, b, (short)0, c, false, false);
    }

    // C/D layout: VGPR r, lanes 0-15 -> M=r,   N=lane
    //                      lanes 16-31 -> M=r+8, N=lane-16
    const int j   = jb + rsub;
    const float sqj = sq[j];
    const int mbase = ib + ((lane >> 4) << 3);
#pragma unroll
    for (int r = 0; r < 8; ++r) {
        const int i = mbase + r;
        float d = sq[i] - 2.0f * c[r] + sqj;
        d = fmaxf(d, 0.0f);
        const float z = (d == 0.0f) ? 1.0f : 0.0f;
        d = __builtin_sqrtf(d + z * EPS_F) * (1.0f - z);
        pd[(size_t)i * N + j] = d;
    }
}

// ---------------------------------------------------------------------------
// Kernel 3: triplet reduction. Block = 256 threads, grid = (N/256, N).
// blockIdx.y = anchor i; thread owns one positive candidate j.
//
// The staged row is PRE-MASKED: srow[k] = pd[i,k] if label[k] != label[i],
// else +INF. Then relu(apm - srow[k]) is automatically 0 for invalid k
// (apm - INF = -INF -> fmax 0; count check false), making the inner loop
// fully branchless: one ds_load_b128 per 4 elements + ~4 VALU ops each.
// Valid triplet mask reduces to: i!=j && lbl[i]==lbl[j] && lbl[k]!=lbl[i]
// (k!=i and k!=j are implied by the label conditions).
// ---------------------------------------------------------------------------
__global__ __launch_bounds__(256) void triplet_kernel(
        const float* __restrict__ pd, const int* __restrict__ labels,
        float* __restrict__ sum_acc, unsigned int* __restrict__ cnt_acc,
        int N) {
    __shared__ __align__(16) float srow[512];   // masked row: +INF at invalid k
    __shared__ float red_s[256];
    __shared__ unsigned int red_c[256];

    const int tid = threadIdx.x;
    const int i   = blockIdx.y;
    const int j   = blockIdx.x * blockDim.x + tid;
    const int li  = labels[i];                  // block-uniform

    for (int k = tid; k < N; k += blockDim.x) {
        const float v = pd[(size_t)i * N + k];
        srow[k] = (labels[k] != li) ? v : __builtin_inff();
    }
    __syncthreads();

    float sum = 0.0f;
    unsigned int cnt = 0u;

    if (j < N && j != i && labels[j] == li) {
        // srow[j] is masked to INF (lbl[j]==li), so read pd[i,j] from global
        const float apm = pd[(size_t)i * N + j] + MARGIN;
        const float4* s4 = (const float4*)srow;
#pragma unroll 4
        for (int k4 = 0; k4 < N / 4; ++k4) {
            const float4 s = s4[k4];
            const float v0 = fmaxf(apm - s.x, 0.0f);
            const float v1 = fmaxf(apm - s.y, 0.0f);
            const float v2 = fmaxf(apm - s.z, 0.0f);
            const float v3 = fmaxf(apm - s.w, 0.0f);
            sum += (v0 + v1) + (v2 + v3);
            cnt += (unsigned)(v0 > EPS_F) + (unsigned)(v1 > EPS_F) +
                   (unsigned)(v2 > EPS_F) + (unsigned)(v3 > EPS_F);
        }
    }

    red_s[tid] = sum;
    red_c[tid] = cnt;
    __syncthreads();
    for (int off = 128; off > 0; off >>= 1) {
        if (tid < off) {
            red_s[tid] += red_s[tid + off];
            red_c[tid] += red_c[tid + off];
        }
        __syncthreads();
    }
    if (tid == 0) {
        atomicAdd(sum_acc, red_s[0]);
        atomicAdd(cnt_acc, red_c[0]);
    }
}

// ---------------------------------------------------------------------------
// Kernel 4: finalize  loss = sum / (num_hard + eps) ; out = [loss, num_hard]
// ---------------------------------------------------------------------------
__global__ void finalize_kernel(const float* sum_acc,
                                const unsigned int* cnt_acc, float* out) {
    if (threadIdx.x == 0) {
        const float nh = (float)(*cnt_acc);
        out[0] = (*sum_acc) / (nh + EPS_F);
        out[1] = nh;
    }
}

extern "C" void kernel_launch(void* const* d_in, const int* in_sizes, int n_in,
                              void* d_out, int out_size, void* d_ws, size_t ws_size,
                              hipStream_t stream) {
    const float* E      = (const float*)d_in[0];
    const int*   labels = (const int*)d_in[1];
    float*       out    = (float*)d_out;

    const int N = in_sizes[1];        // 512
    const int D = in_sizes[0] / N;    // 256

    // Workspace layout: pd[N*N] | sq[N] | sum(float) | cnt(u32)
    float*        pd      = (float*)d_ws;
    float*        sq      = pd + (size_t)N * N;
    float*        sum_acc = sq + N;
    unsigned int* cnt_acc = (unsigned int*)(sum_acc + 1);

    init_acc_kernel<<<1, 32, 0, stream>>>(sum_acc, cnt_acc);

    rownorm_kernel<<<(N + 255) / 256, 256, 0, stream>>>(E, sq, N, D);

    dim3 gGrid(N / 16, N / 16);
    gram_pd_kernel<<<gGrid, 32, 0, stream>>>(E, sq, pd, N, D);

    dim3 tGrid((N + 255) / 256, N);
    triplet_kernel<<<tGrid, 256, 0, stream>>>(pd, labels, sum_acc, cnt_acc, N);

    finalize_kernel<<<1, 32, 0, stream>>>(sum_acc, cnt_acc, out);
}